// Decoder_80642305949765
// MI455X (gfx1250) — compile-verified
//
#include <hip/hip_runtime.h>

typedef __attribute__((ext_vector_type(2))) float v2f;
typedef __attribute__((ext_vector_type(8))) float v8f;

#define HOP      32
#define BATCH    8
#define NDIM     512            // reduction dim (n)
#define KDIM     32768          // frames (k)
#define LDIM     64             // frame length (l)
#define TOUT     ((KDIM - 1) * HOP + LDIM)   // 1048608
#define KCHUNK   256            // k per workgroup
#define NTHREADS 256            // 8 waves x 32 k-columns each
#define FSTRIDE  69             // padded frames row stride (conflict-free)

__device__ __forceinline__ v8f wmma4(v2f a, v2f b, v8f c) {
    return __builtin_amdgcn_wmma_f32_16x16x4_f32(false, a, false, b, (short)0, c, false, false);
}

__global__ __launch_bounds__(NTHREADS)
void decoder_wmma_f32(const float* __restrict__ c,
                      const float* __restrict__ V,
                      float* __restrict__ out) {
    // V pre-swizzled into WMMA-A fragment order: [nstep][tile][lane]{2f}
    __shared__ float Aswz[LDIM * NDIM];             // 131072 B
    __shared__ float Fl[(KCHUNK + 1) * FSTRIDE];    //  70932 B (row 0 = frame K0-1)
    __shared__ float colbuf[NDIM];                  //   2048 B

    const int tid  = threadIdx.x;
    const int lane = tid & 31;
    const int wid  = tid >> 5;                      // 0..7
    const int b    = blockIdx.y;
    const int K0   = blockIdx.x * KCHUNK;

    const float* cb = c + (size_t)b * NDIM * KDIM;

    // ---- stage V[l][n] into A-fragment layout (one-time swizzle) ----
    // element V[l][n] -> slot ((n>>2)*4 + (l>>4))*64 + (l&15)*2 + ((n>>1)&1)*32 + (n&1)
    for (int idx = tid; idx < LDIM * NDIM; idx += NTHREADS) {
        int l = idx >> 9, n = idx & (NDIM - 1);
        int slot = (((n >> 2) * 4 + (l >> 4)) * 32 + ((l & 15) + ((n >> 1) & 1) * 16)) * 2 + (n & 1);
        Aswz[slot] = V[idx];
    }

    // ---- boundary frame (K0-1), upper half l=32..63 -> Fl row 0 ----
    if (tid < 32) Fl[32 + tid] = 0.0f;
    if (K0 != 0) {
        colbuf[tid]       = cb[(size_t)tid * KDIM + (K0 - 1)];
        colbuf[tid + 256] = cb[(size_t)(tid + 256) * KDIM + (K0 - 1)];
    }
    __syncthreads();

    if (K0 != 0) {
        // 256 threads: thread -> (l = 32 + tid/8, 64 n-values), reduce via ds_add_f32
        int l   = 32 + (tid >> 3);
        int n0  = (tid & 7) * 64;
        int t   = l >> 4, jin2 = (l & 15) * 2;
        float s = 0.0f;
        #pragma unroll
        for (int i = 0; i < 64; ++i) {
            int n = n0 + i;
            s += Aswz[(n >> 2) * 256 + t * 64 + jin2 + ((n >> 1) & 1) * 32 + (n & 1)] * colbuf[n];
        }
        atomicAdd(&Fl[l], s);
    }

    // ---- main GEMM: frames[l=0..63][k = k0..k0+31] per wave ----
    const int jin = lane & 15;
    const int hi  = lane >> 4;                      // 0: n rows {0,1} | 1: {2,3}
    const int k0  = K0 + wid * 32;

    const float* bp  = cb + (size_t)(hi * 2) * KDIM + (size_t)(k0 + jin);
    const float* ap0 = &Aswz[lane * 2];

    v8f acc[8];                                     // [kt 0..1][lt 0..3]
    #pragma unroll
    for (int i = 0; i < 8; ++i) acc[i] = (v8f){};

    #pragma unroll 4
    for (int nstep = 0; nstep < NDIM / 4; ++nstep) {
        v2f b0, b1;
        b0.x = bp[0];        b0.y = bp[KDIM];       // k-cols k0..k0+15
        b1.x = bp[16];       b1.y = bp[KDIM + 16];  // k-cols k0+16..k0+31
        bp  += (size_t)4 * KDIM;

        const float* ap = ap0 + nstep * 256;
        v2f a0 = *(const v2f*)(ap);                 // l-tile 0..15
        v2f a1 = *(const v2f*)(ap + 64);            // 16..31
        v2f a2 = *(const v2f*)(ap + 128);           // 32..47
        v2f a3 = *(const v2f*)(ap + 192);           // 48..63

        acc[0] = wmma4(a0, b0, acc[0]);
        acc[1] = wmma4(a1, b0, acc[1]);
        acc[2] = wmma4(a2, b0, acc[2]);
        acc[3] = wmma4(a3, b0, acc[3]);
        acc[4] = wmma4(a0, b1, acc[4]);
        acc[5] = wmma4(a1, b1, acc[5]);
        acc[6] = wmma4(a2, b1, acc[6]);
        acc[7] = wmma4(a3, b1, acc[7]);
    }

    // ---- scatter frames to LDS: row = k - K0 + 1, col = l ----
    #pragma unroll
    for (int kt = 0; kt < 2; ++kt) {
        float* fp = &Fl[(k0 + 16 * kt + jin - K0 + 1) * FSTRIDE + hi * 8];
        #pragma unroll
        for (int lt = 0; lt < 4; ++lt) {
            #pragma unroll
            for (int i = 0; i < 8; ++i)
                fp[16 * lt + i] = acc[kt * 4 + lt][i];   // l = 16*lt + i + 8*hi
        }
    }
    __syncthreads();

    // ---- overlap-add + coalesced float4 stores:
    // y[b, 32q+r] = frames[q][r] + frames[q-1][r+32]
    float* ob = out + (size_t)b * TOUT + (size_t)K0 * HOP;
    float4* ob4 = (float4*)ob;
    #pragma unroll
    for (int i = 0; i < 8; ++i) {
        int t4 = tid + i * NTHREADS;                // float4 index 0..2047
        int tl = t4 * 4;
        int q  = tl >> 5;
        int r  = tl & 31;                            // r..r+3 stay within one frame
        const float* f1 = &Fl[(q + 1) * FSTRIDE + r];
        const float* f0 = &Fl[q * FSTRIDE + 32 + r];
        float4 v;
        v.x = f1[0] + f0[0];
        v.y = f1[1] + f0[1];
        v.z = f1[2] + f0[2];
        v.w = f1[3] + f0[3];
        ob4[t4] = v;
    }
    // tail: last block emits y[32*KDIM .. 32*KDIM+31] = frames[KDIM-1][32..63]
    if (blockIdx.x == gridDim.x - 1 && tid < 32) {
        ob[KCHUNK * HOP + tid] = Fl[KCHUNK * FSTRIDE + 32 + tid];
    }
}

extern "C" void kernel_launch(void* const* d_in, const int* in_sizes, int n_in,
                              void* d_out, int out_size, void* d_ws, size_t ws_size,
                              hipStream_t stream) {
    const float* c  = (const float*)d_in[0];   // [8, 512, 32768] f32
    const float* V  = (const float*)d_in[1];   // [64, 512]       f32
    float*      out = (float*)d_out;           // [8, 1, 1048608] f32

    dim3 grid(KDIM / KCHUNK, BATCH);           // 128 x 8
    decoder_wmma_f32<<<grid, NTHREADS, 0, stream>>>(c, V, out);
}